// QuadrotorDynamics_29772713296343
// MI455X (gfx1250) — compile-verified
//
#include <hip/hip_runtime.h>

typedef float v2f __attribute__((ext_vector_type(2)));
typedef float v8f __attribute__((ext_vector_type(8)));

#define EPS_BN 1e-5f

__global__ __launch_bounds__(256) void quad_dyn_kernel(
    const float* __restrict__ x,        // (1,16,65)
    const float* __restrict__ conv_w,   // (10,10,8)
    const float* __restrict__ conv_b,   // (10,)
    const float* __restrict__ bn_gamma, // (10,)
    const float* __restrict__ bn_beta,  // (10,)
    const float* __restrict__ bn_mean,  // (10,)
    const float* __restrict__ bn_var,   // (10,)
    const float* __restrict__ fc1_w,    // (128,640)
    const float* __restrict__ fc1_b,    // (128,)
    const float* __restrict__ fc2_w,    // (6,132)
    const float* __restrict__ fc2_b,    // (6,)
    float* __restrict__ out)            // (1,16,65)
{
    __shared__ float sx[640];                    // x_init = x[:,6:16,0:64], channel-major
    __shared__ float s1[640];                    // after conv1+bn+relu
    __shared__ __align__(16) float sh[640];      // flattened h (8B-aligned for b64 loads)
    __shared__ float sw[800];                    // conv weights
    __shared__ float sscale[10], soff[10];
    __shared__ float h2[132];                    // fc1 out (128) ++ ff (4)
    __shared__ float corr[6];
    __shared__ float sdot[16];

    const int tid = threadIdx.x;

    // ---- Stage inputs into LDS; prefetch fc1_w into L2 while convs run ----
    for (int i = tid; i < 640; i += 256) {
        int c = i >> 6, t = i & 63;
        sx[i] = x[(6 + c) * 65 + t];
    }
    for (int i = tid; i < 800; i += 256) sw[i] = conv_w[i];
    if (tid < 10) {
        float sc = bn_gamma[tid] * rsqrtf(bn_var[tid] + EPS_BN);
        sscale[tid] = sc;
        soff[tid]   = (conv_b[tid] - bn_mean[tid]) * sc + bn_beta[tid];
    }
    if (tid < 4) h2[128 + tid] = x[(12 + tid) * 65 + 64];   // ff = x[:,12:,-1]
    for (int i = tid * 32; i < 128 * 640; i += 256 * 32)
        __builtin_prefetch(&fc1_w[i], 0, 1);
    __syncthreads();

    // ---- conv1 (dilated causal) + bn + relu ----
    for (int i = tid; i < 640; i += 256) {
        int c = i >> 6, t = i & 63;
        float acc = 0.f;
        const float* wr = &sw[c * 80];
        for (int ci = 0; ci < 10; ++ci) {
            const float* xr = &sx[ci * 64];
            #pragma unroll
            for (int k = 0; k < 8; ++k) {
                int src = t + 2 * k - 14;
                acc += wr[ci * 8 + k] * ((src >= 0) ? xr[src] : 0.f);
            }
        }
        float v = acc * sscale[c] + soff[c];
        s1[i] = v > 0.f ? v : 0.f;
    }
    __syncthreads();

    // ---- conv2 + bn + residual + relu -> flattened h (c*64 + t order) ----
    for (int i = tid; i < 640; i += 256) {
        int c = i >> 6, t = i & 63;
        float acc = 0.f;
        const float* wr = &sw[c * 80];
        for (int ci = 0; ci < 10; ++ci) {
            const float* xr = &s1[ci * 64];
            #pragma unroll
            for (int k = 0; k < 8; ++k) {
                int src = t + 2 * k - 14;
                acc += wr[ci * 8 + k] * ((src >= 0) ? xr[src] : 0.f);
            }
        }
        float v = acc * sscale[c] + soff[c] + sx[i];
        sh[i] = v > 0.f ? v : 0.f;
    }
    __syncthreads();

    // ---- fc1: (1x640)@(640x128)^T via V_WMMA_F32_16X16X4_F32 ----
    // 8 waves; wave w owns output columns [16w, 16w+16). M padded 1->16 by
    // zero-masking A rows 1..15 (mask-multiply: no divergence, no EXEC games).
    {
        const int wave = tid >> 5;       // 0..7
        const int lane = tid & 31;
        const int row  = lane & 15;      // M for A / N for B
        const int half = lane >> 4;      // K-pair selector {0,1} vs {2,3}
        const int n    = wave * 16 + row;
        const float amask = (row == 0) ? 1.f : 0.f;
        const float* wrow = &fc1_w[n * 640];
        v8f acc = {};
        for (int k = 0; k < 640; k += 4) {
            // A: uniform LDS broadcast (ds_load_b64), masked to row M=0 only
            v2f hv = *(const v2f*)&sh[k + 2 * half];
            v2f a;
            a.x = hv.x * amask;
            a.y = hv.y * amask;
            // B[k][n] = fc1_w[n][k]  (global_load_b64, 8B aligned)
            v2f b = *(const v2f*)&wrow[k + 2 * half];
            acc = __builtin_amdgcn_wmma_f32_16x16x4_f32(
                false, a, false, b, (short)0, acc, false, false);
        }
        // D layout: VGPR0, lanes 0-15 -> M=0, N=lane
        if (half == 0) {
            float v = acc[0] + fc1_b[n];
            h2[n] = v > 0.f ? v : 0.f;
        }
    }
    __syncthreads();

    // ---- fc2: (1x132)@(132x6)^T ----
    if (tid < 6) {
        float acc = fc2_b[tid];
        const float* wr = &fc2_w[tid * 132];
        for (int i = 0; i < 132; ++i) acc += wr[i] * h2[i];
        corr[tid] = acc;
    }
    __syncthreads();

    // ---- rigid-body physics (scalar, lane 0) ----
    if (tid == 0) {
        float st[16];
        for (int c = 0; c < 16; ++c) st[c] = x[c * 65 + 63];   // x[:,:,-2]

        const float TH[4][3] = {{ 0.0011f, -0.0069f, 2.2929f},
                                {-0.0005f, -0.0088f, 2.5556f},
                                { 0.0010f, -0.0121f, 2.2989f},
                                {-0.0001f, -0.0116f, 2.5572f}};
        float th[4];
        for (int i = 0; i < 4; ++i) {
            float c = st[12 + i];
            th[i] = TH[i][0] * c * c + TH[i][1] * c + TH[i][2];
        }
        const float LA = 0.707f * 0.2f;
        float tq0 = th[0] + th[1] + th[2] + th[3];
        float tq1 = LA * ( th[0] - th[1] - th[2] + th[3]);
        float tq2 = LA * (-th[0] - th[1] + th[2] + th[3]);
        float tq3 = 0.01f * (-th[0] + th[1] - th[2] + th[3]);

        float sphi = sinf(st[0]), cphi = cosf(st[0]);
        float sth  = sinf(st[1]), cth  = cosf(st[1]);
        float spsi = sinf(st[2]), cpsi = cosf(st[2]);

        // rbi column 2 (body z-axis in inertial frame)
        float r02 = cphi * cpsi * sth + sphi * spsi;
        float r12 = cphi * spsi * sth - sphi * cpsi;
        float r22 = cth * cphi;
        float tz  = tq0 / 0.5f;                       // /MASS

        float vel[3]  = {st[9], st[10], st[11]};
        float vdot[3] = { r02 * tz - 0.1f * vel[0],
                          r12 * tz - 0.1f * vel[1],
                          r22 * tz - 0.1f * vel[2] - 9.8067f };

        float rate[3] = {st[6], st[7], st[8]};
        // Euler-rate matrix M and its inverse (cofactor form)
        float m00 = 1.f,  m01 = 0.f,   m02 = -sphi;
        float m10 = 0.f,  m11 = cphi,  m12 = sphi * cth;
        float m20 = 0.f,  m21 = -sphi, m22 = cth * cphi;
        float det = m00 * (m11 * m22 - m12 * m21)
                  - m01 * (m10 * m22 - m12 * m20)
                  + m02 * (m10 * m21 - m11 * m20);
        float id = 1.f / det;
        float i00 = (m11*m22 - m12*m21)*id, i01 = (m02*m21 - m01*m22)*id, i02 = (m01*m12 - m02*m11)*id;
        float i10 = (m12*m20 - m10*m22)*id, i11 = (m00*m22 - m02*m20)*id, i12 = (m02*m10 - m00*m12)*id;
        float i20 = (m10*m21 - m11*m20)*id, i21 = (m01*m20 - m00*m21)*id, i22 = (m00*m11 - m01*m10)*id;
        float adot[3] = { i00*rate[0] + i01*rate[1] + i02*rate[2],
                          i10*rate[0] + i11*rate[1] + i12*rate[2],
                          i20*rate[0] + i21*rate[1] + i22*rate[2] };

        float Iw[3] = {0.0023f * rate[0], 0.0023f * rate[1], 0.004f * rate[2]};
        float cr[3] = { rate[1]*Iw[2] - rate[2]*Iw[1],
                        rate[2]*Iw[0] - rate[0]*Iw[2],
                        rate[0]*Iw[1] - rate[1]*Iw[0] };
        float rdot[3] = { (tq1 - cr[0] - 0.1f * rate[0]) / 0.0023f,
                          (tq2 - cr[1] - 0.1f * rate[1]) / 0.0023f,
                          (tq3 - cr[2] - 0.1f * rate[2]) / 0.004f };

        sdot[0] = adot[0]; sdot[1] = adot[1]; sdot[2] = adot[2];
        sdot[3] = vel[0];  sdot[4] = vel[1];  sdot[5] = vel[2];
        sdot[6] = rdot[0] + corr[0]; sdot[7]  = rdot[1] + corr[1]; sdot[8]  = rdot[2] + corr[2];
        sdot[9] = vdot[0] + corr[3]; sdot[10] = vdot[1] + corr[4]; sdot[11] = vdot[2] + corr[5];
        sdot[12] = 0.f; sdot[13] = 0.f; sdot[14] = 0.f; sdot[15] = 0.f;
    }
    __syncthreads();

    // ---- scatter: zeros everywhere except column -2 ----
    for (int i = tid; i < 16 * 65; i += 256) {
        int c = i / 65, t = i - c * 65;
        out[i] = (t == 63) ? sdot[c] : 0.f;
    }
}

extern "C" void kernel_launch(void* const* d_in, const int* in_sizes, int n_in,
                              void* d_out, int out_size, void* d_ws, size_t ws_size,
                              hipStream_t stream) {
    (void)in_sizes; (void)n_in; (void)out_size; (void)d_ws; (void)ws_size;
    // setup_inputs order: t, x, conv_w, conv_b, bn_gamma, bn_beta, bn_mean,
    //                     bn_var, fc1_w, fc1_b, fc2_w, fc2_b
    const float* x        = (const float*)d_in[1];
    const float* conv_w   = (const float*)d_in[2];
    const float* conv_b   = (const float*)d_in[3];
    const float* bn_gamma = (const float*)d_in[4];
    const float* bn_beta  = (const float*)d_in[5];
    const float* bn_mean  = (const float*)d_in[6];
    const float* bn_var   = (const float*)d_in[7];
    const float* fc1_w    = (const float*)d_in[8];
    const float* fc1_b    = (const float*)d_in[9];
    const float* fc2_w    = (const float*)d_in[10];
    const float* fc2_b    = (const float*)d_in[11];
    float* out = (float*)d_out;

    quad_dyn_kernel<<<1, 256, 0, stream>>>(
        x, conv_w, conv_b, bn_gamma, bn_beta, bn_mean, bn_var,
        fc1_w, fc1_b, fc2_w, fc2_b, out);
}